// DynamicFilterLayer_58780922413140
// MI455X (gfx1250) — compile-verified
//
#include <hip/hip_runtime.h>
#include <stdint.h>

// Dynamic filter layer: out[n,c,h,w] = sum_{i,j} xpad[n,c,h+i-4,w+j-4]*F[n,i*9+j,h,w] + B[n,0,h,w]
// N=2 C=8 H=256 W=512 K=9 pad=4.  fp32.  Memory-bound (AI ~3.3 FLOP/B).

typedef __attribute__((ext_vector_type(4))) float v4f;
typedef __attribute__((ext_vector_type(4))) int   v4i;

#define AS1 __attribute__((address_space(1)))
#define AS3 __attribute__((address_space(3)))

#define N_  2
#define C_  8
#define H_  256
#define W_  512
#define K_  9
#define P_  4
#define TH  16            // tile height (pixels)
#define TW  64            // tile width  (pixels)
#define RW  4             // pixels per thread along W
#define TGX (TW / RW)     // 16 thread-groups along W
#define TY  (TH + 2 * P_) // 24
#define TX  (TW + 2 * P_) // 72
#define TILE_DW (C_ * TY * TX) // 13824 dwords = 54 KB LDS
#define NVEC (TILE_DW / 4)     // 3456 float4 copies

// ---- CDNA5 async global->LDS copy (ASYNCcnt path), with compiling fallbacks ----

__device__ __forceinline__ void async_cp_b128(const float* g, float* l) {
#if __has_builtin(__builtin_amdgcn_global_load_async_to_lds_b128)
  __builtin_amdgcn_global_load_async_to_lds_b128(
      (AS1 v4i*)(v4i*)(void*)(uintptr_t)(const void*)g,
      (AS3 v4i*)(v4i*)(void*)l, 0, 0);
#else
  *(v4f*)l = *(const v4f*)g; // sync fallback: global_load_b128 + ds_store_b128
#endif
}

__device__ __forceinline__ void async_cp_b32(const float* g, float* l) {
#if __has_builtin(__builtin_amdgcn_global_load_async_to_lds_b32)
  __builtin_amdgcn_global_load_async_to_lds_b32(
      (AS1 int*)(int*)(void*)(uintptr_t)(const void*)g,
      (AS3 int*)(int*)(void*)l, 0, 0);
#else
  *l = *g;
#endif
}

__device__ __forceinline__ void wait_async0() {
#if __has_builtin(__builtin_amdgcn_s_wait_asynccnt)
  __builtin_amdgcn_s_wait_asynccnt(0);
#else
  asm volatile("s_wait_asynccnt 0" ::: "memory");
#endif
}

__global__ __launch_bounds__(256) void dyn_filter_kernel(
    const float* __restrict__ x, const float* __restrict__ flt,
    const float* __restrict__ bias, float* __restrict__ out) {
  __shared__ __align__(16) float tile[TILE_DW]; // [c][y][x], x contiguous

  const int tid = threadIdx.x;
  const int n   = blockIdx.z;
  const int h0  = blockIdx.y * TH;
  const int w0  = blockIdx.x * TW;
  const int gh0 = h0 - P_;
  const int gw0 = w0 - P_;
  const float* __restrict__ xb = x + n * (C_ * H_ * W_);

  const bool interior = (blockIdx.y > 0) && (blockIdx.y < gridDim.y - 1) &&
                        (blockIdx.x > 0) && (blockIdx.x < gridDim.x - 1);

  // ---- Stage x halo tile into LDS via async copies ----
  if (interior) {
#pragma unroll
    for (int k = 0; k < (NVEC + 255) / 256; ++k) {
      const int idx = tid + k * 256;
      if (idx < NVEC) {
        const int xv  = idx % (TX / 4);  // 0..17 (float4 column)
        const int rem = idx / (TX / 4);
        const int y   = rem % TY;
        const int c   = rem / TY;
        const float* src = xb + ((c * H_ + (gh0 + y)) * W_ + (gw0 + xv * 4));
        async_cp_b128(src, &tile[(c * TY + y) * TX + xv * 4]);
      }
    }
  } else {
    // Border: per-dword with zero-fill for out-of-bounds (each dword has one writer).
#pragma unroll 2
    for (int k = 0; k < TILE_DW / 256; ++k) { // 54 iters, exact
      const int idx = tid + k * 256;
      const int xx  = idx % TX;
      const int rem = idx / TX;
      const int y   = rem % TY;
      const int c   = rem / TY;
      const int gh  = gh0 + y;
      const int gw  = gw0 + xx;
      if ((unsigned)gh < (unsigned)H_ && (unsigned)gw < (unsigned)W_) {
        async_cp_b32(xb + (c * H_ + gh) * W_ + gw, &tile[idx]);
      } else {
        tile[idx] = 0.0f;
      }
    }
  }
  wait_async0();     // drain this wave's ASYNCcnt before the barrier
  __syncthreads();   // (compiler also drains DScnt here)

  // ---- Compute: each thread -> 4 W-adjacent pixels x 8 channels ----
  const int lxg = tid & (TGX - 1);
  const int ly  = tid >> 4;
  const int h   = h0 + ly;
  const int w   = w0 + lxg * RW;

  const v4f bv = __builtin_nontemporal_load((const v4f*)(bias + (n * H_ + h) * W_ + w));
  v4f acc[C_];
#pragma unroll
  for (int c = 0; c < C_; ++c) acc[c] = bv;

  const int fbase = (n * (K_ * K_)) * (H_ * W_) + h * W_ + w;

#pragma unroll 1
  for (int i = 0; i < K_; ++i) {
    // Filter row: 9 taps x 4 pixels, streaming (zero reuse -> non-temporal).
    v4f frow[K_];
#pragma unroll
    for (int j = 0; j < K_; ++j)
      frow[j] = __builtin_nontemporal_load(
          (const v4f*)(flt + fbase + (i * K_ + j) * (H_ * W_)));

    // Two channel-halves to cap register pressure (f row reused for both).
#pragma unroll
    for (int half = 0; half < 2; ++half) {
      v4f xav[4][3]; // 12-float sliding window per channel (16B-aligned LDS loads)
#pragma unroll
      for (int cc = 0; cc < 4; ++cc) {
        const float* row = &tile[((half * 4 + cc) * TY + (ly + i)) * TX + lxg * RW];
        xav[cc][0] = *(const v4f*)(row);
        xav[cc][1] = *(const v4f*)(row + 4);
        xav[cc][2] = *(const v4f*)(row + 8);
      }
#pragma unroll
      for (int j = 0; j < K_; ++j) {
#pragma unroll
        for (int cc = 0; cc < 4; ++cc) {
          const int c = half * 4 + cc;
          acc[c].x = fmaf(xav[cc][(j + 0) >> 2][(j + 0) & 3], frow[j].x, acc[c].x);
          acc[c].y = fmaf(xav[cc][(j + 1) >> 2][(j + 1) & 3], frow[j].y, acc[c].y);
          acc[c].z = fmaf(xav[cc][(j + 2) >> 2][(j + 2) & 3], frow[j].z, acc[c].z);
          acc[c].w = fmaf(xav[cc][(j + 3) >> 2][(j + 3) & 3], frow[j].w, acc[c].w);
        }
      }
    }
  }

#pragma unroll
  for (int c = 0; c < C_; ++c) {
    __builtin_nontemporal_store(acc[c], (v4f*)(out + ((n * C_ + c) * H_ + h) * W_ + w));
  }
}

extern "C" void kernel_launch(void* const* d_in, const int* in_sizes, int n_in,
                              void* d_out, int out_size, void* d_ws, size_t ws_size,
                              hipStream_t stream) {
  (void)in_sizes; (void)n_in; (void)out_size; (void)d_ws; (void)ws_size;
  const float* x    = (const float*)d_in[0]; // x_in          [2,8,256,512]
  const float* flt  = (const float*)d_in[1]; // filters       [2,81,256,512]
  const float* bias = (const float*)d_in[2]; // filters_biases[2,1,256,512]
  float* out        = (float*)d_out;         // [2,8,256,512]

  dim3 grid(W_ / TW, H_ / TH, N_); // (8,16,2)
  dim3 block(256);
  hipLaunchKernelGGL(dyn_filter_kernel, grid, block, 0, stream, x, flt, bias, out);
}